// AffineTransformerBlock_43052752175300
// MI455X (gfx1250) — compile-verified
//
#include <hip/hip_runtime.h>

// float2 as a native clang vector so arithmetic maps to packed VALU ops
typedef __attribute__((ext_vector_type(2))) float v2f;

#define BD 256
// Fixed problem shape from the reference: B=8, D=H=W=128, C=2
// strides (in floats): d-dim = 128*128*2 = 1<<15, h-dim = 128*2 = 1<<8, w-dim = 2 = 1<<1

__global__ __launch_bounds__(BD) void affine_trilerp_kernel(
    const float* __restrict__ img,   // (8,128,128,128,2) f32
    const float* __restrict__ tmat,  // (8,3,4) f32
    float* __restrict__ out,         // (8,128,128,128,2) f32
    int nvox)                        // B*D*H*W
{
    const int idx = blockIdx.x * BD + threadIdx.x;
    if (idx >= nvox) return;

    // idx -> (b, d, h, w); lanes run along w for contiguous gathers
    const int wI = idx & 127;
    const int hI = (idx >> 7) & 127;
    const int dI = (idx >> 14) & 127;
    // batch is uniform per wave (2^21 voxels per batch, block = 256 threads):
    // readfirstlane puts the matrix fetch + theta math on the scalar path.
    const int bI = __builtin_amdgcn_readfirstlane(idx >> 21);

    const float* m = tmat + bI * 12;
    const float t   = m[3] * 0.2f;                       // reference uses only m[0,3]
    const float off = fmaf(128.0f, t + 0.5f, -0.5f);     // dims*(t+0.5) - 0.5, dims==128
    const float t00 = fmaf(m[0],  0.2f, 1.0f), t01 = m[1] * 0.2f,            t02 = m[2]  * 0.2f;
    const float t10 = m[4] * 0.2f,             t11 = fmaf(m[5], 0.2f, 1.0f), t12 = m[6]  * 0.2f;
    const float t20 = m[8] * 0.2f,             t21 = m[9] * 0.2f,            t22 = fmaf(m[10], 0.2f, 1.0f);

    // centered grid coords: arange(k) - k/2 - 0.5 with k = 128
    const float gx = (float)dI - 64.5f;
    const float gy = (float)hI - 64.5f;
    const float gz = (float)wI - 64.5f;

    // ref_j = sum_k theta[j,k]*g_k + off   (9 FMAs)
    const float rx = fmaf(t00, gx, fmaf(t01, gy, fmaf(t02, gz, off)));
    const float ry = fmaf(t10, gx, fmaf(t11, gy, fmaf(t12, gz, off)));
    const float rz = fmaf(t20, gx, fmaf(t21, gy, fmaf(t22, gz, off)));

    // init = floor(ref) in one op each
    const int ix = __float2int_rd(rx);
    const int iy = __float2int_rd(ry);
    const int iz = __float2int_rd(rz);

    // clamped corner indices (reference clamps loc = clip(init+s, 0, dim-1))
    const int x0 = min(max(ix,     0), 127), x1 = min(max(ix + 1, 0), 127);
    const int y0 = min(max(iy,     0), 127), y1 = min(max(iy + 1, 0), 127);
    const int z0 = min(max(iz,     0), 127), z1 = min(max(iz + 1, 0), 127);

    // per-dim corner weights: max(1 - |ref - loc|, 0) on the CLAMPED loc
    const float wx0 = fmaxf(1.0f - fabsf(rx - (float)x0), 0.0f);
    const float wx1 = fmaxf(1.0f - fabsf(rx - (float)x1), 0.0f);
    const float wy0 = fmaxf(1.0f - fabsf(ry - (float)y0), 0.0f);
    const float wy1 = fmaxf(1.0f - fabsf(ry - (float)y1), 0.0f);
    const float wz0 = fmaxf(1.0f - fabsf(rz - (float)z0), 0.0f);
    const float wz1 = fmaxf(1.0f - fabsf(rz - (float)z1), 0.0f);

    // element offsets (floats) — all power-of-two shifts
    const float* base = img + ((size_t)bI << 22);        // b * 128^3 * 2
    const int xo0 = x0 << 15, xo1 = x1 << 15;
    const int yo0 = y0 << 8,  yo1 = y1 << 8;
    const int zo0 = z0 << 1,  zo1 = z1 << 1;

    // issue all 8 float2 gathers (global_load_b64) before consuming — these
    // hit L2 (entire 134MB input is resident in the 192MB L2)
    const v2f c000 = *(const v2f*)(base + xo0 + yo0 + zo0);
    const v2f c001 = *(const v2f*)(base + xo0 + yo0 + zo1);
    const v2f c010 = *(const v2f*)(base + xo0 + yo1 + zo0);
    const v2f c011 = *(const v2f*)(base + xo0 + yo1 + zo1);
    const v2f c100 = *(const v2f*)(base + xo1 + yo0 + zo0);
    const v2f c101 = *(const v2f*)(base + xo1 + yo0 + zo1);
    const v2f c110 = *(const v2f*)(base + xo1 + yo1 + zo0);
    const v2f c111 = *(const v2f*)(base + xo1 + yo1 + zo1);

    const float wxy00 = wx0 * wy0, wxy01 = wx0 * wy1;
    const float wxy10 = wx1 * wy0, wxy11 = wx1 * wy1;

    v2f acc = c000 * (wxy00 * wz0);
    acc += c001 * (wxy00 * wz1);
    acc += c010 * (wxy01 * wz0);
    acc += c011 * (wxy01 * wz1);
    acc += c100 * (wxy10 * wz0);
    acc += c101 * (wxy10 * wz1);
    acc += c110 * (wxy11 * wz0);
    acc += c111 * (wxy11 * wz1);

    // Non-temporal store (th:TH_STORE_NT): output is write-once/stream —
    // keep it from evicting the L2-resident input volume.
    __builtin_nontemporal_store(acc, (v2f*)(out + ((size_t)idx << 1)));
}

extern "C" void kernel_launch(void* const* d_in, const int* in_sizes, int n_in,
                              void* d_out, int out_size, void* d_ws, size_t ws_size,
                              hipStream_t stream) {
    const float* img  = (const float*)d_in[0];   // images  (8,128,128,128,2)
    const float* tmat = (const float*)d_in[1];   // trans_mats (8,3,4)
    float* out = (float*)d_out;

    const int nvox = in_sizes[0] / 2;            // B*D*H*W = 16,777,216
    const int blocks = (nvox + BD - 1) / BD;     // 65,536 blocks of 256 (8 waves/WGP-dispatch)

    affine_trilerp_kernel<<<blocks, BD, 0, stream>>>(img, tmat, out, nvox);
}